// Matrix_P_update_82162724372630
// MI455X (gfx1250) — compile-verified
//
#include <hip/hip_runtime.h>
#include <hip/hip_bf16.h>
#include <math.h>

// ---------------------------------------------------------------------------
// Shapes: B=4, C=32, C2=64, H=W=256, HW=65536, Wf=129, SPEC_N=256*129=33024
// All matmul-shaped stages use V_WMMA_F32_16X16X4_F32 (fp32 in/out, wave32).
// ---------------------------------------------------------------------------

typedef float v2f __attribute__((ext_vector_type(2)));
typedef float v8f __attribute__((ext_vector_type(8)));

__device__ __forceinline__ v8f wmma4(v2f a, v2f b, v8f c) {
  // 8-arg pattern: (neg_a, A, neg_b, B, c_mod, C, reuse_a, reuse_b)
  return __builtin_amdgcn_wmma_f32_16x16x4_f32(false, a, false, b, (short)0, c,
                                               false, false);
}

#define HW   65536
#define SPN  33024   // 256*129

// ---------------------------------------------------------------------------
// Generic 256x256x256 fp32 WMMA GEMM. mode 0: m = u@q (into X layout).
// mode 1: P = m1 @ m2^T (B transposed via strides), C -> d_out.
// Block = 128 thr (4 waves), block tile 64x64, K chunk 16 via LDS.
// ---------------------------------------------------------------------------
__global__ __launch_bounds__(128) void gemm256_kernel(
    int mode, const float* __restrict__ P0, const float* __restrict__ P1,
    float* __restrict__ P2) {
  __shared__ float As[64][17];
  __shared__ float Bs[16][65];
  const int z = blockIdx.z;
  const int m0 = blockIdx.y * 64, n0 = blockIdx.x * 64;
  const float* A;
  const float* Bm;
  float* Cm;
  size_t b_rs, b_cs;
  if (mode == 0) {                        // A=u[b,c], B=q[b,c], C=X[b,c]
    A = P0 + (size_t)z * HW;
    Bm = P1 + (size_t)z * HW;
    Cm = P2 + ((size_t)(z >> 5) * 64 + (size_t)(z & 31)) * HW;
    b_rs = 256; b_cs = 1;
  } else {                                // A=out[b,c], B=out[b,32+c]^T
    size_t base = ((size_t)(z >> 5) * 64 + (size_t)(z & 31)) * HW;
    A = P0 + base;
    Bm = P0 + base + (size_t)32 * HW;
    Cm = P2 + (size_t)z * HW;
    b_rs = 1; b_cs = 256;                 // B(k,n) = m2[n,k]
  }
  const int tid = threadIdx.x, wave = tid >> 5, lane = tid & 31;
  const int lh = lane >> 4, ll = lane & 15;
  v8f acc[4] = {};
  for (int kb = 0; kb < 256; kb += 16) {
    for (int f = tid; f < 1024; f += 128) {
      int k = f & 15, m = f >> 4;
      As[m][k] = A[(size_t)(m0 + m) * 256 + (size_t)(kb + k)];
    }
    for (int f = tid; f < 1024; f += 128) {
      int n = f & 63, k = f >> 6;
      Bs[k][n] = Bm[(size_t)(kb + k) * b_rs + (size_t)(n0 + n) * b_cs];
    }
    if (kb + 16 < 256)  // hint next A chunk -> global_prefetch_b8
      __builtin_prefetch(&A[(size_t)(m0 + (tid >> 4)) * 256 + kb + 16], 0, 3);
    __syncthreads();
    const int ml = wave * 16 + ll;
#pragma unroll
    for (int kk = 0; kk < 16; kk += 4) {
      v2f a;
      a.x = As[ml][kk + 2 * lh];
      a.y = As[ml][kk + 2 * lh + 1];
#pragma unroll
      for (int nt = 0; nt < 4; ++nt) {
        v2f bv;
        bv.x = Bs[kk + 2 * lh][nt * 16 + ll];
        bv.y = Bs[kk + 2 * lh + 1][nt * 16 + ll];
        acc[nt] = wmma4(a, bv, acc[nt]);
      }
    }
    __syncthreads();
  }
#pragma unroll
  for (int nt = 0; nt < 4; ++nt) {
    int n = n0 + nt * 16 + ll;
#pragma unroll
    for (int r = 0; r < 8; ++r) {
      int m = m0 + wave * 16 + lh * 8 + r;
      Cm[(size_t)m * 256 + n] = acc[nt][r];
    }
  }
}

// ---------------------------------------------------------------------------
// X[:,32:64] = z
// ---------------------------------------------------------------------------
__global__ void copyz_kernel(const float* __restrict__ z, float* __restrict__ X) {
  size_t i = (size_t)blockIdx.x * 256 + threadIdx.x;
  size_t b = i / (size_t)(32 * HW);
  size_t rem = i - b * (size_t)(32 * HW);
  X[b * (size_t)(64 * HW) + (size_t)(32 * HW) + rem] = z[i];
}

// ---------------------------------------------------------------------------
// Per-(b,c2) sum / sumsq over HxW -> partial[bc2*2 + {0,1}]
// ---------------------------------------------------------------------------
__global__ __launch_bounds__(256) void stats_kernel(const float* __restrict__ X,
                                                    float* __restrict__ partial) {
  __shared__ float s1[256], s2[256];
  const float* p = X + (size_t)blockIdx.x * HW;
  float a = 0.f, qq = 0.f;
  for (int i = threadIdx.x; i < HW; i += 256) {
    float v = p[i];
    a += v; qq += v * v;
  }
  s1[threadIdx.x] = a; s2[threadIdx.x] = qq;
  __syncthreads();
  for (int s = 128; s > 0; s >>= 1) {
    if (threadIdx.x < s) {
      s1[threadIdx.x] += s1[threadIdx.x + s];
      s2[threadIdx.x] += s2[threadIdx.x + s];
    }
    __syncthreads();
  }
  if (threadIdx.x == 0) {
    partial[blockIdx.x * 2] = s1[0];
    partial[blockIdx.x * 2 + 1] = s2[0];
  }
}

// ---------------------------------------------------------------------------
// Channel-attention MLP (64->21 relu, 21->64 sigmoid) + BN affine coeffs.
// ---------------------------------------------------------------------------
__global__ __launch_bounds__(64) void prep_kernel(
    const float* __restrict__ partial, const float* __restrict__ ca_w1,
    const float* __restrict__ ca_b1, const float* __restrict__ ca_w2,
    const float* __restrict__ ca_b2, const float* __restrict__ gamma,
    const float* __restrict__ beta, float* __restrict__ s_out,
    float* __restrict__ bnA, float* __restrict__ bnB) {
  __shared__ float g[4][64];
  __shared__ float hid[4][21];
  const int t = threadIdx.x;
  for (int i = t; i < 256; i += 64) g[i >> 6][i & 63] = partial[i * 2] * (1.0f / 65536.0f);
  __syncthreads();
  for (int i = t; i < 84; i += 64) {
    int b = i / 21, o = i % 21;
    float acc = ca_b1[o];
    for (int c = 0; c < 64; ++c) acc += g[b][c] * ca_w1[o * 64 + c];
    hid[b][o] = fmaxf(acc, 0.f);
  }
  __syncthreads();
  for (int i = t; i < 256; i += 64) {
    int b = i >> 6, o = i & 63;
    float acc = ca_b2[o];
    for (int c = 0; c < 21; ++c) acc += hid[b][c] * ca_w2[o * 21 + c];
    s_out[i] = 1.0f / (1.0f + expf(-acc));
  }
  // BatchNorm (population stats over b,h,w)
  float sum = 0.f, sq = 0.f;
  for (int b = 0; b < 4; ++b) {
    sum += partial[(b * 64 + t) * 2];
    sq += partial[(b * 64 + t) * 2 + 1];
  }
  float mu = sum * (1.0f / 262144.0f);
  float var = sq * (1.0f / 262144.0f) - mu * mu;
  float Af = gamma[t] * rsqrtf(var + 1e-5f);
  bnA[t] = Af;
  bnB[t] = beta[t] - mu * Af;
}

// ---------------------------------------------------------------------------
// 3x3 'SAME' conv as implicit GEMM via WMMA: out[co,pix] += W_tap[co,ci] *
// Xshift[ci,pix], 9 taps, K chunked by 32 ci. Block tile: 64co x 64pix (one
// image row segment). Zero-padded 66-wide row slab staged in LDS.
// ---------------------------------------------------------------------------
__global__ __launch_bounds__(128) void conv3_kernel(
    const float* __restrict__ in, const float* __restrict__ wgt,
    float* __restrict__ out, const float* __restrict__ scale,
    float* __restrict__ ymax, int relu) {
  __shared__ float Tsh[32][198];  // [ci][r*66 + w], rows h-1..h+1, zero padded
  __shared__ float Ash[64][33];   // [co][ci] for current tap
  __shared__ float wmax[8][64];
  const int b = blockIdx.z, h = blockIdx.y, wb = blockIdx.x * 64;
  const int tid = threadIdx.x, wave = tid >> 5, lane = tid & 31;
  const int lh = lane >> 4, ll = lane & 15;
  v8f acc[4] = {};
  for (int cic = 0; cic < 2; ++cic) {
    const int ci0 = cic * 32;
    __syncthreads();  // protect Tsh from previous chunk's readers
    for (int f = tid; f < 32 * 198; f += 128) {
      int w66 = f % 66, r = (f / 66) % 3, ci = f / 198;
      int gw = wb - 1 + w66, gh = h - 1 + r;
      float v = 0.f;
      if (gw >= 0 && gw < 256 && gh >= 0 && gh < 256) {
        v = in[(((size_t)b * 64 + ci0 + ci) * 256 + gh) * 256 + gw];
        if (scale) v *= scale[b * 64 + ci0 + ci];   // fused SE scaling (conv1)
      }
      Tsh[ci][r * 66 + w66] = v;
    }
    for (int tap = 0; tap < 9; ++tap) {
      const int kh = tap / 3, kw = tap - kh * 3;
      __syncthreads();  // Tsh staged / prev tap done
      for (int f = tid; f < 2048; f += 128) {
        int ci = f & 31, co = f >> 5;
        Ash[co][ci] = wgt[(((size_t)co * 64 + ci0 + ci) * 3 + kh) * 3 + kw];
      }
      __syncthreads();
      const int ml = wave * 16 + ll;
#pragma unroll
      for (int kk = 0; kk < 32; kk += 4) {
        v2f a;
        a.x = Ash[ml][kk + 2 * lh];
        a.y = Ash[ml][kk + 2 * lh + 1];
#pragma unroll
        for (int nt = 0; nt < 4; ++nt) {
          int n = nt * 16 + ll;
          v2f bv;
          bv.x = Tsh[kk + 2 * lh][kh * 66 + kw + n];
          bv.y = Tsh[kk + 2 * lh + 1][kh * 66 + kw + n];
          acc[nt] = wmma4(a, bv, acc[nt]);
        }
      }
    }
  }
#pragma unroll
  for (int nt = 0; nt < 4; ++nt) {
    int n = nt * 16 + ll;
    float mx = -1e30f;
#pragma unroll
    for (int r = 0; r < 8; ++r) {
      float v = acc[nt][r];
      if (relu) v = fmaxf(v, 0.f);
      int co = wave * 16 + lh * 8 + r;
      out[(((size_t)b * 64 + co) * 256 + h) * 256 + wb + n] = v;
      mx = fmaxf(mx, v);
    }
    if (ymax) wmax[wave * 2 + lh][n] = mx;
  }
  if (ymax) {  // y1_max = max(max_co y1, 1.0), fused (block owns all 64 co)
    __syncthreads();
    if (tid < 64) {
      float m = 1.0f;
#pragma unroll
      for (int i = 0; i < 8; ++i) m = fmaxf(m, wmax[i][tid]);
      ymax[(size_t)b * HW + (size_t)h * 256 + wb + tid] = m;
    }
  }
}

// ---------------------------------------------------------------------------
// Radix-2 LDS FFT, 256 points, 128 threads (one butterfly each per stage).
// ---------------------------------------------------------------------------
__device__ void fft256_lds(float* re, float* im, int t, float sign) {
  for (int i = t; i < 256; i += 128) {
    int j = (int)(__brev((unsigned)i) >> 24);
    if (j > i) {
      float a = re[i]; re[i] = re[j]; re[j] = a;
      float b = im[i]; im[i] = im[j]; im[j] = b;
    }
  }
  __syncthreads();
  for (int len = 2; len <= 256; len <<= 1) {
    int half = len >> 1;
    int j = t & (half - 1);
    int i0 = (t / half) * len + j;
    int i1 = i0 + half;
    float ang = sign * 6.283185307179586f * (float)j / (float)len;
    float c = cosf(ang), s = sinf(ang);
    float xr = re[i1], xi = im[i1];
    float tr = xr * c - xi * s;
    float ti = xr * s + xi * c;
    float ur = re[i0], ui = im[i0];
    re[i0] = ur + tr; im[i0] = ui + ti;
    re[i1] = ur - tr; im[i1] = ui - ti;
    __syncthreads();
  }
}

// Forward row rFFT with fused BatchNorm affine. Writes half-spectrum.
__global__ __launch_bounds__(128) void fft_row_fwd_kernel(
    const float* __restrict__ X, const float* __restrict__ bnA,
    const float* __restrict__ bnB, float* __restrict__ SR,
    float* __restrict__ SI, int b) {
  __shared__ float re[256], im[256];
  const int h = blockIdx.x, c2 = blockIdx.y, t = threadIdx.x;
  const float* row = X + (((size_t)b * 64 + c2) * 256 + h) * 256;
  const float Af = bnA[c2], Bf = bnB[c2];
  for (int i = t; i < 256; i += 128) { re[i] = row[i] * Af + Bf; im[i] = 0.f; }
  __syncthreads();
  fft256_lds(re, im, t, -1.0f);
  for (int i = t; i < 129; i += 128) {
    SR[(size_t)c2 * SPN + (size_t)h * 129 + i] = re[i];
    SI[(size_t)c2 * SPN + (size_t)h * 129 + i] = im[i];
  }
}

// Complex FFT along H (stride-129 columns), fwd (sign=-1) or inv (sign=+1).
__global__ __launch_bounds__(128) void fft_col_kernel(float* __restrict__ SR,
                                                      float* __restrict__ SI,
                                                      float sign) {
  __shared__ float re[256], im[256];
  const int wf = blockIdx.x, c2 = blockIdx.y, t = threadIdx.x;
  const size_t base = (size_t)c2 * SPN + wf;
  for (int i = t; i < 256; i += 128) {
    re[i] = SR[base + (size_t)i * 129];
    im[i] = SI[base + (size_t)i * 129];
  }
  __syncthreads();
  fft256_lds(re, im, t, sign);
  for (int i = t; i < 256; i += 128) {
    SR[base + (size_t)i * 129] = re[i];
    SI[base + (size_t)i * 129] = im[i];
  }
}

// ---------------------------------------------------------------------------
// Spectral channel MLP: C[o,n] = s1*sum_k W1[k,o]*B1[k,n]
//                               + s2*sum_k W2[k,o]*B2[k,n]   (opt. ReLU)
// M=64, K=64, N=33024. Signs folded into A-fragment staging (f32 WMMA has
// only C-negate). Block 64x64 tile, dual WMMA per k-step.
// ---------------------------------------------------------------------------
__global__ __launch_bounds__(128) void mlp_gemm_kernel(
    const float* __restrict__ W1, float s1, const float* __restrict__ B1,
    const float* __restrict__ W2, float s2, const float* __restrict__ B2,
    float* __restrict__ Cm, int relu) {
  __shared__ float A1s[64][17], A2s[64][17];
  __shared__ float B1s[16][65], B2s[16][65];
  const int n0 = blockIdx.x * 64;
  const int tid = threadIdx.x, wave = tid >> 5, lane = tid & 31;
  const int lh = lane >> 4, ll = lane & 15;
  v8f acc[4] = {};
  for (int kb = 0; kb < 64; kb += 16) {
    for (int f = tid; f < 1024; f += 128) {
      int k = f & 15, o = f >> 4;
      A1s[o][k] = s1 * W1[(size_t)(kb + k) * 64 + o];  // transposed weight read
      A2s[o][k] = s2 * W2[(size_t)(kb + k) * 64 + o];
    }
    for (int f = tid; f < 1024; f += 128) {
      int n = f & 63, k = f >> 6;
      B1s[k][n] = B1[(size_t)(kb + k) * SPN + n0 + n];
      B2s[k][n] = B2[(size_t)(kb + k) * SPN + n0 + n];
    }
    __syncthreads();
    const int ml = wave * 16 + ll;
#pragma unroll
    for (int kk = 0; kk < 16; kk += 4) {
      v2f a1, a2;
      a1.x = A1s[ml][kk + 2 * lh]; a1.y = A1s[ml][kk + 2 * lh + 1];
      a2.x = A2s[ml][kk + 2 * lh]; a2.y = A2s[ml][kk + 2 * lh + 1];
#pragma unroll
      for (int nt = 0; nt < 4; ++nt) {
        v2f b1, b2;
        b1.x = B1s[kk + 2 * lh][nt * 16 + ll];
        b1.y = B1s[kk + 2 * lh + 1][nt * 16 + ll];
        b2.x = B2s[kk + 2 * lh][nt * 16 + ll];
        b2.y = B2s[kk + 2 * lh + 1][nt * 16 + ll];
        acc[nt] = wmma4(a1, b1, acc[nt]);
        acc[nt] = wmma4(a2, b2, acc[nt]);
      }
    }
    __syncthreads();
  }
#pragma unroll
  for (int nt = 0; nt < 4; ++nt) {
    int n = n0 + nt * 16 + ll;
#pragma unroll
    for (int r = 0; r < 8; ++r) {
      int o = wave * 16 + lh * 8 + r;
      float v = acc[nt][r];
      if (relu) v = fmaxf(v, 0.f);
      Cm[(size_t)o * SPN + n] = v;
    }
  }
}

// ---------------------------------------------------------------------------
// Inverse row FFT (Hermitian mirror 129->256) + fused tanh-warp and
// out = y1 + y1max*tanh(y2/y1max) + x.
// ---------------------------------------------------------------------------
__global__ __launch_bounds__(128) void fft_row_inv_combine_kernel(
    const float* __restrict__ SR, const float* __restrict__ SI,
    const float* __restrict__ Y1, const float* __restrict__ X,
    const float* __restrict__ Y1MAX, float* __restrict__ OUT, int b) {
  __shared__ float re[256], im[256];
  const int h = blockIdx.x, c2 = blockIdx.y, t = threadIdx.x;
  const size_t sb = (size_t)c2 * SPN + (size_t)h * 129;
  for (int i = t; i < 129; i += 128) { re[i] = SR[sb + i]; im[i] = SI[sb + i]; }
  __syncthreads();
  for (int i = 129 + t; i < 256; i += 128) {  // F[k] = conj(F[256-k])
    re[i] = re[256 - i];
    im[i] = -im[256 - i];
  }
  __syncthreads();
  fft256_lds(re, im, t, 1.0f);
  const size_t rowi = (((size_t)b * 64 + c2) * 256 + h) * 256;
  const size_t ymb = (size_t)b * HW + (size_t)h * 256;
  for (int i = t; i < 256; i += 128) {
    float ys = re[i] * (1.0f / 65536.0f);  // 1/256 per inverse transform
    float ym = Y1MAX[ymb + i];             // >= 1.0, no div-by-zero
    float y2 = ym * tanhf(ys / ym);
    OUT[rowi + i] = Y1[rowi + i] + y2 + X[rowi + i];
  }
}

// ---------------------------------------------------------------------------
// L2-normalize each (b,c) 256x256 slab of P in place.
// ---------------------------------------------------------------------------
__global__ __launch_bounds__(256) void pnorm_kernel(float* __restrict__ P) {
  __shared__ float sh[256];
  __shared__ float inv;
  const size_t base = (size_t)blockIdx.x * HW;
  float qq = 0.f;
  for (int i = threadIdx.x; i < HW; i += 256) {
    float v = P[base + i];
    qq += v * v;
  }
  sh[threadIdx.x] = qq;
  __syncthreads();
  for (int s = 128; s > 0; s >>= 1) {
    if (threadIdx.x < s) sh[threadIdx.x] += sh[threadIdx.x + s];
    __syncthreads();
  }
  if (threadIdx.x == 0) inv = 1.0f / fmaxf(sqrtf(sh[0]), 1e-12f);
  __syncthreads();
  for (int i = threadIdx.x; i < HW; i += 256) P[base + i] *= inv;
}

// ---------------------------------------------------------------------------
extern "C" void kernel_launch(void* const* d_in, const int* in_sizes, int n_in,
                              void* d_out, int out_size, void* d_ws,
                              size_t ws_size, hipStream_t stream) {
  (void)in_sizes; (void)n_in; (void)out_size; (void)ws_size;
  const float* u = (const float*)d_in[0];
  const float* z = (const float*)d_in[1];
  const float* q = (const float*)d_in[2];
  const float* conv1_w = (const float*)d_in[3];
  const float* conv2_w = (const float*)d_in[4];
  const float* ca_w1 = (const float*)d_in[5];
  const float* ca_b1 = (const float*)d_in[6];
  const float* ca_w2 = (const float*)d_in[7];
  const float* ca_b2 = (const float*)d_in[8];
  const float* w1 = (const float*)d_in[9];    // [2,64,64]
  const float* w2 = (const float*)d_in[10];   // [2,64,64]
  const float* gamma = (const float*)d_in[11];
  const float* beta = (const float*)d_in[12];
  float* out = (float*)d_out;
  float* ws = (float*)d_ws;

  // Workspace layout (floats); ~236 MB total.
  float* X = ws;                       // [4,64,256,256] concat(m,z)
  float* Y1 = ws + 16777216;           // conv branch output
  float* T = ws + 2 * 16777216;        // conv temp, then final 'out' tensor
  float* SR = ws + 3 * 16777216;       // per-batch spectrum real [64,33024]
  float* SI = SR + 2113536;
  float* AR = SI + 2113536;            // layer-1 activations
  float* AI = AR + 2113536;
  float* YMAX = AI + 2113536;          // [4,256,256]
  float* STAT = YMAX + 262144;
  float* Sca = STAT;                   // SE scales [4,64]
  float* bnA = STAT + 256;             // gamma*rsqrt(var+eps)
  float* bnB = STAT + 320;             // beta - mu*bnA
  float* part = STAT + 384;            // per-(b,c2) sum/sumsq [256*2]

  // 1) x = concat(u@q, z)
  gemm256_kernel<<<dim3(4, 4, 128), 128, 0, stream>>>(0, u, q, X);
  copyz_kernel<<<dim3(32768), 256, 0, stream>>>(z, X);

  // 2) SE-gate + BatchNorm statistics
  stats_kernel<<<dim3(256), 256, 0, stream>>>(X, part);
  prep_kernel<<<1, 64, 0, stream>>>(part, ca_w1, ca_b1, ca_w2, ca_b2, gamma,
                                    beta, Sca, bnA, bnB);

  // 3) conv branch: relu(conv1(x*s)) -> conv2 -> y1, y1max
  conv3_kernel<<<dim3(4, 256, 4), 128, 0, stream>>>(X, conv1_w, T, Sca,
                                                    nullptr, 1);
  conv3_kernel<<<dim3(4, 256, 4), 128, 0, stream>>>(T, conv2_w, Y1, nullptr,
                                                    YMAX, 0);

  // 4) spectral branch, per batch (reuses spectrum scratch; stream-serial)
  for (int b = 0; b < 4; ++b) {
    fft_row_fwd_kernel<<<dim3(256, 64), 128, 0, stream>>>(X, bnA, bnB, SR, SI, b);
    fft_col_kernel<<<dim3(129, 64), 128, 0, stream>>>(SR, SI, -1.0f);
    // ar = relu(fr@w1r - fi@w1i), ai = relu(fr@w1i + fi@w1r)
    mlp_gemm_kernel<<<dim3(516), 128, 0, stream>>>(w1, 1.f, SR, w1 + 4096, -1.f, SI, AR, 1);
    mlp_gemm_kernel<<<dim3(516), 128, 0, stream>>>(w1 + 4096, 1.f, SR, w1, 1.f, SI, AI, 1);
    // br = ar@w2r - ai@w2i, bi = ar@w2i + ai@w2r (write back into SR/SI)
    mlp_gemm_kernel<<<dim3(516), 128, 0, stream>>>(w2, 1.f, AR, w2 + 4096, -1.f, AI, SR, 0);
    mlp_gemm_kernel<<<dim3(516), 128, 0, stream>>>(w2 + 4096, 1.f, AR, w2, 1.f, AI, SI, 0);
    fft_col_kernel<<<dim3(129, 64), 128, 0, stream>>>(SR, SI, 1.0f);
    fft_row_inv_combine_kernel<<<dim3(256, 64), 128, 0, stream>>>(SR, SI, Y1, X,
                                                                  YMAX, T, b);
  }

  // 5) P = m1 @ m2^T per (b,c), then L2-normalize over flattened spatial
  gemm256_kernel<<<dim3(4, 4, 128), 128, 0, stream>>>(1, T, nullptr, out);
  pnorm_kernel<<<dim3(128), 256, 0, stream>>>(out);
}